// ScriptKalmanFilter_18056042512875
// MI455X (gfx1250) — compile-verified
//
#include <hip/hip_runtime.h>

#define G_ 2048
#define T_ 200
#define S_ 8
#define M_ 2

typedef __attribute__((ext_vector_type(2))) float v2f;
typedef __attribute__((ext_vector_type(8))) float v8f;

__device__ __forceinline__ v8f wmma4(v2f a, v2f b, v8f c) {
  // D(16x16,f32) = A(16x4,f32) * B(4x16,f32) + C
  return __builtin_amdgcn_wmma_f32_16x16x4_f32(false, a, false, b, (short)0, c,
                                               false, false);
}

__global__ __launch_bounds__(32) void kf_kernel(
    const float* __restrict__ inp,       // [G,T,M]
    const float* __restrict__ Fm,        // [S,S]
    const float* __restrict__ Hm,        // [M,S]
    const float* __restrict__ Qm,        // [S,S]
    const float* __restrict__ Rm,        // [M,M]
    const float* __restrict__ init_mean, // [G,S]
    const float* __restrict__ init_cov,  // [G,S,S]
    float* __restrict__ out_means,       // [G,T,S]
    float* __restrict__ out_covs)        // [G,T,S,S]
{
  __shared__ float flds[64], qlds[64], hlds[16], rlds[4];
  __shared__ float pbuf[256];   // block-diag covariance 16x16, row-major
  __shared__ float abuf[256];   // staging tile (D-layout -> row-major)
  __shared__ float mean_lds[16];
  __shared__ float pht_lds[32]; // [g][s][j]
  __shared__ float k_lds[32];   // [g][s][j]
  __shared__ float fk_lds[32];  // [g][s][j]  (= -(F@K))
  __shared__ float mu_lds[16];
  __shared__ float res_lds[4];

  const int l    = threadIdx.x;     // 0..31
  const int g0   = blockIdx.x * 2;  // two groups per wave
  const int half = l >> 4;          // lane half selects K-pair / row block
  const int li   = l & 15;

  // ---- cooperative loads of static matrices + initial state ----
  for (int i = l; i < 64; i += 32) { flds[i] = Fm[i]; qlds[i] = Qm[i]; }
  if (l < 16) hlds[l] = Hm[l];
  if (l < 4)  rlds[l] = Rm[l];
  if (l < 16) mean_lds[l] = init_mean[(size_t)g0 * S_ + l]; // groups adjacent
  for (int i = l; i < 256; i += 32) pbuf[i] = 0.0f;
  __syncthreads();
  { // scatter init covariances into block-diagonal positions
    int i0 = l * 4;
    int g = i0 >> 6, r = (i0 >> 3) & 7, c0 = i0 & 7;
#pragma unroll
    for (int q = 0; q < 4; ++q)
      pbuf[(8*g + r) * 16 + 8*g + c0 + q] =
          init_cov[(size_t)(g0 + g) * 64 + r * 8 + c0 + q];
  }
  __syncthreads();

  // ---- constant WMMA fragments ----
  // C/D-layout fragments of blockdiag(F,F) and blockdiag(Q,Q)
  v8f fbC, qbC;
#pragma unroll
  for (int v = 0; v < 8; ++v) {
    int row = v + half * 8, col = li;
    float fv = 0.f, qv = 0.f;
    if ((row < 8) == (col < 8)) {
      int rr = row & 7, cc = col & 7;
      fv = flds[rr * 8 + cc];
      qv = qlds[rr * 8 + cc];
    }
    fbC[v] = fv; qbC[v] = qv;
  }
  // B-layout (4x16) fragment of Hb = [[H,0],[0,H]] (rows 0-1 g0, rows 2-3 g1)
  v2f hbB;
  {
    int n = li;
    if (!half) {
      hbB[0] = (n < 8) ? hlds[0 * 8 + n] : 0.f;
      hbB[1] = (n < 8) ? hlds[1 * 8 + n] : 0.f;
    } else {
      hbB[0] = (n >= 8) ? hlds[0 * 8 + (n - 8)] : 0.f;
      hbB[1] = (n >= 8) ? hlds[1 * 8 + (n - 8)] : 0.f;
    }
  }
  // B-layout K-chunks of blockdiag(F^T, F^T):  FbT(k,n) = Fb(n,k)
  v2f fbtB[4];
#pragma unroll
  for (int kc = 0; kc < 4; ++kc) {
    int k0 = 4 * kc + half * 2;  // k0 even -> k0,k0+1 in same 8-block
    int n = li;
    float bx = 0.f, by = 0.f;
    if ((n < 8) == (k0 < 8)) {
      bx = flds[(n & 7) * 8 + (k0 & 7)];
      by = flds[(n & 7) * 8 + ((k0 + 1) & 7)];
    }
    fbtB[kc][0] = bx; fbtB[kc][1] = by;
  }

  const int gA = half;            // group handled by this lane for 32-way jobs
  const int sA = li >> 1;
  const int jA = li & 1;

  for (int t = 0; t < T_; ++t) {
    // ---- 1. emit prediction (carry at start of step) ----
    if (l < 16) {
      int g = l >> 3, s = l & 7;
      out_means[((size_t)(g0 + g) * T_ + t) * S_ + s] = mean_lds[l];
    }
    {
      int i0 = l * 4;
      int g = i0 >> 6, r = (i0 >> 3) & 7, c0 = i0 & 7;
      const float* src = &pbuf[(8*g + r) * 16 + 8*g + c0];
      float4 val = make_float4(src[0], src[1], src[2], src[3]);
      *(float4*)&out_covs[((size_t)(g0 + g) * T_ + t) * 64 + r * 8 + c0] = val;
    }
    // ---- 2. PHt (one elem/lane) + residual ----
    float y0 = inp[((size_t)(g0 + gA) * T_ + t) * M_ + 0];
    float y1 = inp[((size_t)(g0 + gA) * T_ + t) * M_ + 1];
    {
      float acc = 0.f;
      const float* prow = &pbuf[(8*gA + sA) * 16 + 8*gA];
#pragma unroll
      for (int k = 0; k < 8; ++k) acc += prow[k] * hlds[jA * 8 + k];
      pht_lds[gA * 16 + sA * 2 + jA] = acc;
      float r0 = y0, r1 = y1;
#pragma unroll
      for (int k = 0; k < 8; ++k) {
        float mk = mean_lds[gA * 8 + k];
        r0 -= hlds[k] * mk;
        r1 -= hlds[8 + k] * mk;
      }
      if (li == 0) { res_lds[gA * 2 + 0] = r0; res_lds[gA * 2 + 1] = r1; }
    }
    __syncthreads();
    // ---- 3. S = H PHt + R, analytic 2x2 inverse, K = PHt S^-1 ----
    {
      float s00 = rlds[0], s01 = rlds[1], s10 = rlds[2], s11 = rlds[3];
#pragma unroll
      for (int k = 0; k < 8; ++k) {
        float p0 = pht_lds[gA * 16 + k * 2 + 0];
        float p1 = pht_lds[gA * 16 + k * 2 + 1];
        s00 += hlds[k] * p0;      s01 += hlds[k] * p1;
        s10 += hlds[8 + k] * p0;  s11 += hlds[8 + k] * p1;
      }
      float inv = 1.0f / (s00 * s11 - s01 * s10);
      float i00 =  s11 * inv, i01 = -s01 * inv;
      float i10 = -s10 * inv, i11 =  s00 * inv;
      float kv = pht_lds[gA * 16 + sA * 2 + 0] * (jA ? i01 : i00)
               + pht_lds[gA * 16 + sA * 2 + 1] * (jA ? i11 : i10);
      k_lds[gA * 16 + sA * 2 + jA] = kv;
    }
    __syncthreads();
    // ---- 4. mean update; FKn = -(F@K) ----
    if (l < 16) {
      int g = l >> 3, s = l & 7;
      mu_lds[l] = mean_lds[l]
                + k_lds[g * 16 + s * 2 + 0] * res_lds[g * 2 + 0]
                + k_lds[g * 16 + s * 2 + 1] * res_lds[g * 2 + 1];
    }
    {
      float acc = 0.f;
#pragma unroll
      for (int k = 0; k < 8; ++k)
        acc += flds[sA * 8 + k] * k_lds[gA * 16 + k * 2 + jA];
      fk_lds[gA * 16 + sA * 2 + jA] = -acc;
    }
    __syncthreads();
    // ---- 5. mean predict; WMMA: Ab = (-FK)b @ Hb + Fb  (= F(I-KH) blockdiag)
    float mp = 0.f;
    if (l < 16) {
      int g = l >> 3, s = l & 7;
#pragma unroll
      for (int k = 0; k < 8; ++k) mp += flds[s * 8 + k] * mu_lds[g * 8 + k];
    }
    v2f afrag;  // A-layout 16x4: blockdiag cols 0-1 = g0 FK, cols 2-3 = g1 FK
    {
      int m = li;
      if (!half) {
        afrag[0] = (m < 8) ? fk_lds[m * 2 + 0] : 0.f;
        afrag[1] = (m < 8) ? fk_lds[m * 2 + 1] : 0.f;
      } else {
        afrag[0] = (m >= 8) ? fk_lds[16 + (m - 8) * 2 + 0] : 0.f;
        afrag[1] = (m >= 8) ? fk_lds[16 + (m - 8) * 2 + 1] : 0.f;
      }
    }
    v8f ab = wmma4(afrag, hbB, fbC);
    __syncthreads();
    if (l < 16) mean_lds[l] = mp;
#pragma unroll
    for (int v = 0; v < 8; ++v) abuf[(v + half * 8) * 16 + li] = ab[v];
    __syncthreads();
    // ---- 6. T1 = Ab @ Pb : 4 chained K=4 WMMAs ----
    v8f t1;
#pragma unroll
    for (int v = 0; v < 8; ++v) t1[v] = 0.f;
#pragma unroll
    for (int kc = 0; kc < 4; ++kc) {
      int kb = 4 * kc + half * 2;
      v2f a; a[0] = abuf[li * 16 + kb];  a[1] = abuf[li * 16 + kb + 1];
      v2f b; b[0] = pbuf[kb * 16 + li];  b[1] = pbuf[(kb + 1) * 16 + li];
      t1 = wmma4(a, b, t1);
    }
    __syncthreads();
#pragma unroll
    for (int v = 0; v < 8; ++v) abuf[(v + half * 8) * 16 + li] = t1[v];
    __syncthreads();
    // ---- 7. P' = T1 @ FbT + Qb : 4 chained WMMAs, C seeded with Qb ----
    v8f pn = qbC;
#pragma unroll
    for (int kc = 0; kc < 4; ++kc) {
      int kb = 4 * kc + half * 2;
      v2f a; a[0] = abuf[li * 16 + kb];  a[1] = abuf[li * 16 + kb + 1];
      pn = wmma4(a, fbtB[kc], pn);
    }
    __syncthreads();
#pragma unroll
    for (int v = 0; v < 8; ++v) pbuf[(v + half * 8) * 16 + li] = pn[v];
    __syncthreads();
  }
}

// Broadcast static R [2,2] and H [2,8] to [G,T,...] output slabs.
__global__ void bcast_rh(const float* __restrict__ Rm,
                         const float* __restrict__ Hm,
                         float* __restrict__ outR, float* __restrict__ outH) {
  int tid = blockIdx.x * blockDim.x + threadIdx.x;  // 0..G*T-1
  if (tid >= G_ * T_) return;
  float4 r = make_float4(Rm[0], Rm[1], Rm[2], Rm[3]);
  *(float4*)&outR[(size_t)tid * 4] = r;
#pragma unroll
  for (int i = 0; i < 4; ++i) {
    float4 h = make_float4(Hm[i * 4 + 0], Hm[i * 4 + 1],
                           Hm[i * 4 + 2], Hm[i * 4 + 3]);
    *(float4*)&outH[(size_t)tid * 16 + i * 4] = h;
  }
}

extern "C" void kernel_launch(void* const* d_in, const int* in_sizes, int n_in,
                              void* d_out, int out_size, void* d_ws,
                              size_t ws_size, hipStream_t stream) {
  const float* inp = (const float*)d_in[0];
  const float* F   = (const float*)d_in[1];
  const float* H   = (const float*)d_in[2];
  const float* Q   = (const float*)d_in[3];
  const float* R   = (const float*)d_in[4];
  const float* im  = (const float*)d_in[5];
  const float* ic  = (const float*)d_in[6];
  // d_in[7] = n_step (asserted == 1 in reference)

  float* out = (float*)d_out;
  const size_t GT = (size_t)G_ * T_;
  float* out_means = out;                               // [G,T,S]
  float* out_covs  = out + GT * S_;                     // [G,T,S,S]
  float* out_R     = out + GT * (S_ + S_ * S_);         // [G,T,M,M]
  float* out_H     = out + GT * (S_ + S_ * S_ + M_*M_); // [G,T,M,S]

  kf_kernel<<<G_ / 2, 32, 0, stream>>>(inp, F, H, Q, R, im, ic,
                                       out_means, out_covs);
  bcast_rh<<<(G_ * T_ + 255) / 256, 256, 0, stream>>>(R, H, out_R, out_H);
}